// PureTriadicBrain_39324720562581
// MI455X (gfx1250) — compile-verified
//
#include <hip/hip_runtime.h>
#include <math.h>

// ---------------------------------------------------------------------------
// PureTriadicBrain on MI455X (gfx1250).  B=8, N=80, D=128.
// Memory-bound: ~480 MiB streamed in the weight-update pass => ~21us floor at
// 23.3 TB/s.  GEMMs (Q.K^T and A_new.output) use V_WMMA_F32_16X16X4_F32.
// The update pass uses non-temporal (TH_NT) loads/stores for read-once /
// write-once data so the L2-resident W1..W3 (120 MiB < 192 MB L2, warmed by
// the GEMV pass) are not thrashed.
// ---------------------------------------------------------------------------

typedef __attribute__((ext_vector_type(2))) float v2f;
typedef __attribute__((ext_vector_type(4))) float v4f;
typedef __attribute__((ext_vector_type(8))) float v8f;

#define BB 8
#define NN 80
#define DD 128

__device__ __forceinline__ v8f wmma_f32_16x16x4(v2f a, v2f b, v8f c) {
  // 8 args: (neg_a, A, neg_b, B, c_mod, C, reuse_a, reuse_b)
  return __builtin_amdgcn_wmma_f32_16x16x4_f32(false, a, false, b, (short)0, c,
                                               false, false);
}

// ---------------------------------------------------------------------------
// K1: queries = state.W3, keys = state.W2, raw_pred = state.W1 (per node),
//     prediction nonlinearity, counter.  One block per (b,n), 128 threads.
//     Regular-temporal loads on purpose: warms L2 with W1..W3 for K5.
// ---------------------------------------------------------------------------
__global__ void k_gemv_pred(const float* __restrict__ W1,
                            const float* __restrict__ W2,
                            const float* __restrict__ W3,
                            const float* __restrict__ state,
                            const float* __restrict__ stepc,
                            float* __restrict__ q, float* __restrict__ kb,
                            float* __restrict__ pred,
                            float* __restrict__ counter_out) {
  const int bn = blockIdx.x;   // 0..639
  const int k = threadIdx.x;   // 0..127 (output column)
  __shared__ float s[DD];
  s[k] = state[bn * DD + k];
  __syncthreads();
  const size_t base = (size_t)bn * DD * DD;
  const float* w1 = W1 + base;
  const float* w2 = W2 + base;
  const float* w3 = W3 + base;
  float a1 = 0.f, a2 = 0.f, a3 = 0.f;
#pragma unroll 4
  for (int j = 0; j < DD; ++j) {
    const float sj = s[j];
    a1 = fmaf(sj, w1[j * DD + k], a1);
    a2 = fmaf(sj, w2[j * DD + k], a2);
    a3 = fmaf(sj, w3[j * DD + k], a3);
  }
  q[bn * DD + k] = a3;   // queries from W3
  kb[bn * DD + k] = a2;  // keys from W2
  const float raw = a1 - tanhf(a1) * 1.61803398875f;              // UNTANH=PHI
  pred[bn * DD + k] = tanhf(raw) * 1.8477590650225735f;           // sqrt(2+sqrt2)
  if (bn == 0 && k == 0) counter_out[0] = stepc[0] + 1.0f;
}

// ---------------------------------------------------------------------------
// K2: raw_A[b] = Q[b] * K[b]^T / sqrt(D).  80x80x128 GEMM per batch.
//     One wave per 16x16 tile (5x5 tiles), v_wmma_f32_16x16x4_f32, K=128.
// ---------------------------------------------------------------------------
__global__ void k_scores(const float* __restrict__ q,
                         const float* __restrict__ kb,
                         float* __restrict__ rawA) {
  const int idx = blockIdx.x;        // b*25 + tile
  const int b = idx / 25;
  const int tile = idx % 25;
  const int ti = tile / 5, tj = tile % 5;
  const int lane = threadIdx.x;      // 0..31, all active (EXEC all 1s)
  const int m = lane & 15;
  const int kh = lane >> 4;
  // A frag: A[m][2*kh+v] = Q[ti*16+m][k0+2*kh+v]  -> aligned float2 load
  const float* Qp = q + (size_t)b * NN * DD + (size_t)(ti * 16 + m) * DD + 2 * kh;
  // B frag: B[k][n] = K[tj*16+n][k]; n = lane&15  -> aligned float2 load
  const float* Kp = kb + (size_t)b * NN * DD + (size_t)(tj * 16 + m) * DD + 2 * kh;
  v8f c = {};
#pragma unroll 8
  for (int k0 = 0; k0 < DD; k0 += 4) {
    v2f a = *(const v2f*)(Qp + k0);
    v2f bf = *(const v2f*)(Kp + k0);
    c = wmma_f32_16x16x4(a, bf, c);
  }
  const float scale = 0.08838834764831845f;  // 1/sqrt(128)
  float* out = rawA + (size_t)b * NN * NN;
#pragma unroll
  for (int v = 0; v < 8; ++v) {
    const int row = ti * 16 + v + 8 * kh;   // C/D layout: M = v + 8*(lane>>4)
    const int col = tj * 16 + m;            // N = lane&15
    out[row * NN + col] = c[v] * scale;
  }
}

// ---------------------------------------------------------------------------
// K3: top-5 softmax routing + A_new EMA.  One thread per (b,n) row.
// ---------------------------------------------------------------------------
__global__ void k_route(const float* __restrict__ rawA,
                        const float* __restrict__ A,
                        float* __restrict__ Anew) {
  const int t = blockIdx.x * blockDim.x + threadIdx.x;
  if (t >= BB * NN) return;
  const int b = t / NN, n = t % NN;
  float* dst = Anew + (size_t)b * NN * NN + (size_t)n * NN;
  if (n < 2) {  // A_new[:,0:2,:] = 0
    for (int j = 0; j < NN; ++j) dst[j] = 0.0f;
    return;
  }
  const float* row = rawA + (size_t)b * NN * NN + (size_t)n * NN;
  float v[5];
  int id[5];
#pragma unroll
  for (int i = 0; i < 5; ++i) { v[i] = -INFINITY; id[i] = 0; }
  for (int j = 0; j < NN; ++j) {
    const float x = row[j];
    if (x > v[4]) {  // strict '>' keeps earliest index on ties (jax top_k)
      int p = 4;
      while (p > 0 && x > v[p - 1]) { v[p] = v[p - 1]; id[p] = id[p - 1]; --p; }
      v[p] = x; id[p] = j;
    }
  }
  const float mx = v[0];
  float e[5], sum = 0.f;
#pragma unroll
  for (int i = 0; i < 5; ++i) { e[i] = expf(v[i] - mx); sum += e[i]; }
  const float inv = 1.0f / sum;
  const float* Arow = A + (size_t)b * NN * NN + (size_t)n * NN;
  for (int j = 0; j < NN; ++j) dst[j] = Arow[j] * 0.99f;   // C_MOMENT
#pragma unroll
  for (int i = 0; i < 5; ++i) dst[id[i]] += e[i] * inv * 0.01f;
}

// ---------------------------------------------------------------------------
// K4: total_in[b] = A_new[b] (80x80) * output[b] (80x128).  WMMA, K=80.
// ---------------------------------------------------------------------------
__global__ void k_aggregate(const float* __restrict__ Anew,
                            const float* __restrict__ outp,
                            float* __restrict__ totalin) {
  const int idx = blockIdx.x;  // b*40 + tile
  const int b = idx / 40;
  const int tile = idx % 40;
  const int ti = tile / 8, tj = tile % 8;  // 5 row-tiles x 8 col-tiles
  const int lane = threadIdx.x;
  const int m = lane & 15;
  const int kh = lane >> 4;
  const float* An = Anew + (size_t)b * NN * NN;
  const float* O = outp + (size_t)b * NN * DD;
  v8f c = {};
#pragma unroll 5
  for (int k0 = 0; k0 < NN; k0 += 4) {
    const int kk = k0 + 2 * kh;
    v2f a = *(const v2f*)(An + (size_t)(ti * 16 + m) * NN + kk);  // 8B aligned
    v2f bf;
    bf.x = O[(size_t)kk * DD + tj * 16 + m];        // B[k][n] = output[k][n]
    bf.y = O[(size_t)(kk + 1) * DD + tj * 16 + m];
    c = wmma_f32_16x16x4(a, bf, c);
  }
  float* out = totalin + (size_t)b * NN * DD;
#pragma unroll
  for (int v = 0; v < 8; ++v) {
    const int row = ti * 16 + v + 8 * kh;
    const int col = tj * 16 + m;
    out[row * DD + col] = c[v];
  }
}

// ---------------------------------------------------------------------------
// K5: fused target/error/softmax/Eb/advantage + rank-1 Hebbian weight update.
//     One block per (b,n), 128 threads, b128 streaming of 6x64KB in and
//     6x64KB out per block -> the 480 MiB bandwidth-bound pass.
//     M-loads and all stores are non-temporal (read-once / write-once);
//     W-loads are last-use (served from L2 warmed by K1).
// ---------------------------------------------------------------------------
__global__ void k_update(const float* __restrict__ W1, const float* __restrict__ W2,
                         const float* __restrict__ W3, const float* __restrict__ M1,
                         const float* __restrict__ M2, const float* __restrict__ M3,
                         const float* __restrict__ Ebase,
                         const float* __restrict__ state,
                         const float* __restrict__ totalin,
                         const float* __restrict__ pred,
                         const float* __restrict__ eye,
                         const float* __restrict__ stomach,
                         const float* __restrict__ stepc,
                         float* __restrict__ target_out, float* __restrict__ eb_out,
                         float* __restrict__ W1n, float* __restrict__ W2n,
                         float* __restrict__ W3n, float* __restrict__ M1n,
                         float* __restrict__ M2n, float* __restrict__ M3n) {
  const int bn = blockIdx.x;
  const int b = bn / NN, n = bn % NN;
  const int k = threadIdx.x;  // 0..127
  __shared__ float red[DD];
  __shared__ float coef[DD];
  __shared__ float sst[DD];

  const float p = pred[bn * DD + k];
  float tgt;
  if (n == 0)      tgt = eye[b * DD + k];
  else if (n == 1) tgt = stomach[b * DD + k];
  else             tgt = totalin[bn * DD + k];
  target_out[bn * DD + k] = tgt;
  const float err = p - tgt;

  // softmax(error_signal) over D
  red[k] = err;
  __syncthreads();
  for (int s = 64; s > 0; s >>= 1) {
    if (k < s) red[k] = fmaxf(red[k], red[k + s]);
    __syncthreads();
  }
  const float mx = red[0];
  __syncthreads();
  const float ex = expf(err - mx);
  red[k] = ex;
  __syncthreads();
  for (int s = 64; s > 0; s >>= 1) {
    if (k < s) red[k] += red[k + s];
    __syncthreads();
  }
  const float Ecurr = ex / red[0];

  const float ebase = Ebase[bn * DD + k];
  const float bm = (ebase == 0.0f) ? 1.0f : 0.0f;
  const float eb0 = ebase * (1.0f - bm) + Ecurr * bm;
  const float eb = eb0 * 0.5f + 0.5f * Ecurr;  // EMA_SPEED=0.5
  eb_out[bn * DD + k] = eb;
  const float adv = Ecurr - eb;                        // REW_SENS=1
  const float R = -(adv * rsqrtf(adv * adv + 1e-8f));  // -rms_norm (1-elem axis)
  const float plast = 1.0f - R;
  const float mask = (eb > 0.0f) ? 1.0f : 0.0f;
  coef[k] = -plast * err * mask;  // hebb[r][c] = coef[r] * state[c]
  sst[k] = state[bn * DD + k];
  __syncthreads();

  // travelling gaussian over the LAST dim (column c); fixed per thread
  const float cnt = stepc[0] + 1.0f;
  const float center = fmodf(cnt * 1.5f, 128.0f);  // SPEED=1.5
  const int c0 = (k & 31) * 4;
  float gm[4], s4[4];
#pragma unroll
  for (int t = 0; t < 4; ++t) {
    float diff = fabsf((float)(c0 + t) - center);
    diff = fminf(diff, 128.0f - diff);
    gm[t] = expf(-diff * diff / 0.020001f);  // 2*WIDTH^2 + 1e-6
    s4[t] = sst[c0 + t];
  }

  const size_t base = (size_t)bn * (DD * DD);
  const int rb = k >> 5;  // row-phase 0..3
  for (int it = 0; it < 32; ++it) {
    const int r = it * 4 + rb;
    const size_t off = base + (size_t)r * DD + c0;  // 16B aligned
    // W: last-use reads (L2-resident from K1); M: read-once -> non-temporal.
    const v4f w1 = *(const v4f*)(W1 + off);
    const v4f w2 = *(const v4f*)(W2 + off);
    const v4f w3 = *(const v4f*)(W3 + off);
    const v4f m1 = __builtin_nontemporal_load((const v4f*)(M1 + off));
    const v4f m2 = __builtin_nontemporal_load((const v4f*)(M2 + off));
    const v4f m3 = __builtin_nontemporal_load((const v4f*)(M3 + off));
    const float cr = coef[r];
    v4f o1, o2, o3, n1, n2, n3;
#define UPD(cmp, t)                                                 \
  {                                                                 \
    const float g = cr * s4[t];                                     \
    const float g1 = (g + 0.01f * w3.cmp - 0.01f * w1.cmp) * gm[t]; \
    const float g2 = (g + 0.01f * w1.cmp - 0.01f * w2.cmp) * gm[t]; \
    const float g3 = (g + 0.01f * w2.cmp - 0.01f * w3.cmp) * gm[t]; \
    n1.cmp = m1.cmp * 0.42f + 0.58f * g1;                           \
    n2.cmp = m2.cmp * 0.42f + 0.58f * g2;                           \
    n3.cmp = m3.cmp * 0.42f + 0.58f * g3;                           \
    o1.cmp = w1.cmp + 0.001f * n1.cmp;                              \
    o2.cmp = w2.cmp + 0.001f * n2.cmp;                              \
    o3.cmp = w3.cmp + 0.001f * n3.cmp;                              \
  }
    UPD(x, 0) UPD(y, 1) UPD(z, 2) UPD(w, 3)
#undef UPD
    // write-once streaming outputs: non-temporal stores
    __builtin_nontemporal_store(n1, (v4f*)(M1n + off));
    __builtin_nontemporal_store(n2, (v4f*)(M2n + off));
    __builtin_nontemporal_store(n3, (v4f*)(M3n + off));
    __builtin_nontemporal_store(o1, (v4f*)(W1n + off));
    __builtin_nontemporal_store(o2, (v4f*)(W2n + off));
    __builtin_nontemporal_store(o3, (v4f*)(W3n + off));
  }
}

// ---------------------------------------------------------------------------
extern "C" void kernel_launch(void* const* d_in, const int* in_sizes, int n_in,
                              void* d_out, int out_size, void* d_ws,
                              size_t ws_size, hipStream_t stream) {
  (void)in_sizes; (void)n_in; (void)out_size; (void)ws_size;
  const float* W1 = (const float*)d_in[0];
  const float* W2 = (const float*)d_in[1];
  const float* W3 = (const float*)d_in[2];
  const float* M1 = (const float*)d_in[3];
  const float* M2 = (const float*)d_in[4];
  const float* M3 = (const float*)d_in[5];
  const float* Ebase = (const float*)d_in[6];
  const float* state = (const float*)d_in[7];
  const float* outp = (const float*)d_in[8];
  const float* A = (const float*)d_in[9];
  const float* stepc = (const float*)d_in[10];
  const float* eye = (const float*)d_in[11];
  const float* stomach = (const float*)d_in[12];

  // workspace layout (floats)
  float* ws = (float*)d_ws;
  float* q = ws;                         // B*N*D = 81920
  float* kb = q + 81920;                 // 81920
  float* rawA = kb + 81920;              // B*N*N = 51200
  float* totalin = rawA + 51200;         // 81920

  // output layout (floats), concatenated in reference return order
  float* out = (float*)d_out;
  float* pred = out;                     // 81920
  float* target = pred + 81920;          // 81920
  float* Anew = target + 81920;          // 51200
  float* Eb = Anew + 51200;              // 81920
  float* W1n = Eb + 81920;               // 10485760 each
  float* W2n = W1n + 10485760;
  float* W3n = W2n + 10485760;
  float* M1n = W3n + 10485760;
  float* M2n = M1n + 10485760;
  float* M3n = M2n + 10485760;
  float* counter = M3n + 10485760;       // 1

  k_gemv_pred<<<BB * NN, DD, 0, stream>>>(W1, W2, W3, state, stepc, q, kb,
                                          pred, counter);
  k_scores<<<BB * 25, 32, 0, stream>>>(q, kb, rawA);
  k_route<<<(BB * NN + 127) / 128, 128, 0, stream>>>(rawA, A, Anew);
  k_aggregate<<<BB * 40, 32, 0, stream>>>(Anew, outp, totalin);
  k_update<<<BB * NN, DD, 0, stream>>>(W1, W2, W3, M1, M2, M3, Ebase, state,
                                       totalin, pred, eye, stomach, stepc,
                                       target, Eb, W1n, W2n, W3n, M1n, M2n,
                                       M3n);
}